// HybridFCLQuantum_65481071398215
// MI455X (gfx1250) — compile-verified
//
#include <hip/hip_runtime.h>
#include <math.h>

typedef __attribute__((ext_vector_type(2))) float v2f;
typedef __attribute__((ext_vector_type(8))) float v8f;

// Single wave32 kernel. The reference consumes only q.reshape(-1)[:4], i.e. the
// 4 wire measurements of image0/patch0 -> dependency cone is 4 pixels.
//   z[w] = <Z_w> of U @ RY-product-state(patch0)      (exact fp32 via WMMA)
//   out  = sum_s |prod_w cos((z_w - support[s,w])/2)| * fc_w[s] + fc_b
__global__ __launch_bounds__(32)
void quanv_head_kernel(const float* __restrict__ x,       // [8192,1,28,28]
                       const float* __restrict__ Uc,      // [16,16] complex64 interleaved (re,im)
                       const float* __restrict__ support, // [64,4]
                       const float* __restrict__ fc_w,    // [1,64]
                       const float* __restrict__ fc_b,    // [1]
                       float* __restrict__ out)           // [1]
{
    __shared__ float Ulds[512];   // 16x16 complex64 = 2 KB, interleaved (re,im)

    const int lane = threadIdx.x & 31;
    const int m    = lane & 15;   // A-row / D-column index
    const int hi   = lane >> 4;   // wave half selects K sub-pair / M half

    __builtin_prefetch(support, 0, 0);   // global_prefetch_b8

    // ---- Stage U into LDS via CDNA5 async global->LDS copies (ASYNCcnt path) ----
    {
        unsigned           ldsOff = (unsigned)(size_t)(&Ulds[0]) + (unsigned)lane * 16u;
        unsigned long long gaddr  = (unsigned long long)(const void*)Uc + (unsigned long long)lane * 16ull;
#pragma unroll
        for (int i = 0; i < 4; ++i) {   // 4 x (32 lanes x 16B) = 2 KB
            asm volatile("global_load_async_to_lds_b128 %0, %1, off"
                         :: "v"(ldsOff + (unsigned)i * 512u),
                            "v"(gaddr + (unsigned long long)i * 512ull)
                         : "memory");
        }
        asm volatile("s_wait_asynccnt 0x0" ::: "memory");
    }

    // 2x2 patch (0,0) of image 0, reference pixel order [r,c],[r,c+1],[r+1,c],[r+1,c+1]
    // Arguments are small -> hardware transcendentals (v_sin_f32/v_cos_f32).
    const float p[4] = { x[0], x[1], x[28], x[29] };
    float cw[4], sw[4];
#pragma unroll
    for (int w = 0; w < 4; ++w) { float h = 0.5f * p[w]; cw[w] = __cosf(h); sw[w] = __sinf(h); }

    // Product state over 4 wires, wire 0 = MSB (wave-uniform, lives in SGPRs)
    float st[16];
#pragma unroll
    for (int j = 0; j < 16; ++j) {
        float a0 = ((j >> 3) & 1) ? sw[0] : cw[0];
        float a1 = ((j >> 2) & 1) ? sw[1] : cw[1];
        float a2 = ((j >> 1) & 1) ? sw[2] : cw[2];
        float a3 = ( j       & 1) ? sw[3] : cw[3];
        st[j] = a0 * a1 * a2 * a3;
    }

    // ---- psi_r = Ur*s, psi_i = Ui*s : two chains of 4x V_WMMA_F32_16X16X4_F32 ----
    // A 16x4 f32 layout: lanes0-15 {v0:K0, v1:K1}, lanes16-31 {v0:K2, v1:K3}.
    // B is the state broadcast across all 16 columns (only the K mapping matters).
    v8f dr = {}; v8f di = {};
#pragma unroll
    for (int kb = 0; kb < 16; kb += 4) {
        const int k0 = kb + 2 * hi;
        v2f z0 = *(const v2f*)&Ulds[2 * (m * 16 + k0)];       // (re,im) U[m,k0]
        v2f z1 = *(const v2f*)&Ulds[2 * (m * 16 + k0 + 1)];   // (re,im) U[m,k0+1]
        v2f ar; ar.x = z0.x; ar.y = z1.x;
        v2f ai; ai.x = z0.y; ai.y = z1.y;
        v2f b;  b.x = hi ? st[kb + 2] : st[kb];               // constant indices -> 1 cndmask
                b.y = hi ? st[kb + 3] : st[kb + 1];
        dr = __builtin_amdgcn_wmma_f32_16x16x4_f32(false, ar, false, b, (short)0, dr, false, false);
        di = __builtin_amdgcn_wmma_f32_16x16x4_f32(false, ai, false, b, (short)0, di, false, false);
    }

    // D layout: lane holds rows M = v + 8*hi of column N = m (all columns identical).
    // meas[w] = sum_i signs[i,w] * |psi_i|^2, signs[i,w] = 1 - 2*bit_{3-w}(i)
    float meas[4] = {0.f, 0.f, 0.f, 0.f};
#pragma unroll
    for (int v = 0; v < 8; ++v) {
        float prob = dr[v] * dr[v] + di[v] * di[v];
        const int M = v + 8 * hi;
#pragma unroll
        for (int w = 0; w < 4; ++w)
            meas[w] += (((M >> (3 - w)) & 1) ? -prob : prob);
    }
#pragma unroll
    for (int w = 0; w < 4; ++w)
        meas[w] += __shfl_xor(meas[w], 16, 32);   // combine M=0..7 with M=8..15

    // ---- Head: 64 support rows, 2 per lane, then butterfly reduction ----
    float acc = 0.f;
#pragma unroll
    for (int r = lane; r < 64; r += 32) {
        float prod = 1.f;
#pragma unroll
        for (int w = 0; w < 4; ++w)
            prod *= __cosf((meas[w] - support[r * 4 + w]) * 0.5f);
        acc += fabsf(prod) * fc_w[r];
    }
#pragma unroll
    for (int off = 16; off > 0; off >>= 1)
        acc += __shfl_xor(acc, off, 32);

    if (lane == 0)
        out[0] = acc + fc_b[0];
}

extern "C" void kernel_launch(void* const* d_in, const int* in_sizes, int n_in,
                              void* d_out, int out_size, void* d_ws, size_t ws_size,
                              hipStream_t stream) {
    (void)in_sizes; (void)n_in; (void)d_ws; (void)ws_size; (void)out_size;
    const float* x       = (const float*)d_in[0];
    const float* U       = (const float*)d_in[1]; // complex64 -> interleaved float pairs
    const float* support = (const float*)d_in[2];
    const float* fc_w    = (const float*)d_in[3];
    const float* fc_b    = (const float*)d_in[4];
    float* out           = (float*)d_out;

    quanv_head_kernel<<<dim3(1), dim3(32), 0, stream>>>(x, U, support, fc_w, fc_b, out);
}